// SPGAT_1468878815357
// MI455X (gfx1250) — compile-verified
//
#include <hip/hip_runtime.h>
#include <hip/hip_bf16.h>

// ---------------------------------------------------------------------------
// SPGAT for MI455X (gfx1250): 3-layer GAT, fp32 WMMA GEMMs + async-LDS staging
// ---------------------------------------------------------------------------

typedef __attribute__((ext_vector_type(2))) float v2f;
typedef __attribute__((ext_vector_type(8))) float v8f;
typedef __attribute__((ext_vector_type(4))) int   v4i;

#define TILE_M 64
#define TILE_N 128
#define KCHUNK 32
#define LDA 36              // A row stride (floats): 144B, 16B-aligned rows
#define LDB 132             // B row stride (floats): 528B, 16B-aligned rows
#define A_TILE (TILE_M * LDA)
#define B_TILE (KCHUNK * LDB)

#define AS1 __attribute__((address_space(1)))
#define AS3 __attribute__((address_space(3)))

#if defined(__AMDGCN__) && __has_builtin(__builtin_amdgcn_global_load_async_to_lds_b128)
#define SPGAT_ASYNC 1
#else
#define SPGAT_ASYNC 0
#endif

__device__ __forceinline__ void spgat_wait_async0() {
#if defined(__AMDGCN__)
#if __has_builtin(__builtin_amdgcn_s_wait_asynccnt)
    __builtin_amdgcn_s_wait_asynccnt(0);
#else
    asm volatile("s_wait_asynccnt 0x0" ::: "memory");
#endif
#endif
}

// ---------------------------------------------------------------------------
// Tiled GEMM: C[M,N] = A[M,K] * B[K,N], fp32, via v_wmma_f32_16x16x4_f32.
// Block: 256 threads = 8 waves; block tile 64x128; wave tile 16x64.
// A (64x32) and B (32x128) staged in LDS. With async-to-LDS available, the
// fills bypass VGPRs entirely (GLOBAL_LOAD_ASYNC_TO_LDS_B128 / ASYNCcnt) and
// are double-buffered so chunk i+1 streams in while chunk i runs on the
// matrix pipe. N-edge masking (layer 2, N=47) handled at fill time with a
// block-uniform branch; EXEC stays all-ones around every WMMA.
// ---------------------------------------------------------------------------
__global__ __launch_bounds__(256)
void spgat_gemm_wmma(const float* __restrict__ A, const float* __restrict__ B,
                     float* __restrict__ C, int M, int K, int N) {
#if SPGAT_ASYNC
    __shared__ float lds_a[2 * A_TILE];
    __shared__ float lds_b[2 * B_TILE];
#else
    __shared__ float lds_a[A_TILE];
    __shared__ float lds_b[B_TILE];
#endif

    const int tid   = threadIdx.x;
    const int lane  = tid & 31;
    const int wave  = tid >> 5;
    const int half  = lane >> 4;     // 0: lanes 0-15, 1: lanes 16-31
    const int l15   = lane & 15;
    const int m_sub = wave & 3;      // 4 waves along M
    const int n_sub = wave >> 2;     // 2 waves along N
    const int tile_m = blockIdx.x * TILE_M;
    const int tile_n = blockIdx.y * TILE_N;
    const int wm = tile_m + m_sub * 16;
    const int wn_l = n_sub * 64;                   // local N offset in tile
    const bool full_n = (tile_n + TILE_N) <= N;    // block-uniform

    v8f acc[4] = {};

#if SPGAT_ASYNC
    // ---- double-buffered async global->LDS staging ----
    auto fill = [&](int buf, int k0) {
        float* la = lds_a + buf * A_TILE;
        float* lb = lds_b + buf * B_TILE;
        #pragma unroll
        for (int i = 0; i < 2; ++i) {               // A: 64 rows x 32 k
            int linear = tid + i * 256;             // 0..511
            int row    = linear >> 3;               // 8 float4 per row
            int c4     = (linear & 7) * 4;
            int grow   = tile_m + row;
            if (grow >= M) grow = M - 1;            // clamp; masked at store
            __builtin_amdgcn_global_load_async_to_lds_b128(
                (AS1 v4i*)(A + (size_t)grow * K + k0 + c4),
                (AS3 v4i*)(la + row * LDA + c4), 0, 0);
        }
        if (full_n) {
            #pragma unroll
            for (int i = 0; i < 4; ++i) {           // B: 32 k x 128 n
                int linear = tid + i * 256;         // 0..1023
                int krow   = linear >> 5;           // 32 float4 per row
                int c4     = (linear & 31) * 4;
                __builtin_amdgcn_global_load_async_to_lds_b128(
                    (AS1 v4i*)(B + (size_t)(k0 + krow) * N + tile_n + c4),
                    (AS3 v4i*)(lb + krow * LDB + c4), 0, 0);
            }
        } else {                                    // cold path: layer 2 N=47
            #pragma unroll
            for (int i = 0; i < 4; ++i) {
                int linear = tid + i * 256;
                int krow   = linear >> 5;
                int c4     = (linear & 31) * 4;
                float* p = &lb[krow * LDB + c4];
                #pragma unroll
                for (int j = 0; j < 4; ++j) {
                    int col = tile_n + c4 + j;
                    p[j] = (col < N) ? B[(size_t)(k0 + krow) * N + col] : 0.0f;
                }
            }
        }
    };

    const int nch = K / KCHUNK;
    fill(0, 0);
    for (int i = 0; i < nch; ++i) {
        spgat_wait_async0();                 // this wave's fills landed
        __syncthreads();                     // everyone's fills + prev compute
        if (i + 1 < nch) fill((i + 1) & 1, (i + 1) * KCHUNK);
        const float* la = lds_a + (i & 1) * A_TILE;
        const float* lb = lds_b + (i & 1) * B_TILE;
        #pragma unroll
        for (int kk = 0; kk < KCHUNK; kk += 4) {
            const float* ap = &la[(m_sub * 16 + l15) * LDA + kk + half * 2];
            v2f a; a.x = ap[0]; a.y = ap[1];
            const int kb = kk + half * 2;
            #pragma unroll
            for (int t = 0; t < 4; ++t) {
                const float* bp = &lb[kb * LDB + wn_l + t * 16 + l15];
                v2f b; b.x = bp[0]; b.y = bp[LDB];
                acc[t] = __builtin_amdgcn_wmma_f32_16x16x4_f32(
                    false, a, false, b, (short)0, acc[t], false, false);
            }
        }
    }
#else
    // ---- synchronous fallback (VGPR-staged fills) ----
    for (int k0 = 0; k0 < K; k0 += KCHUNK) {
        #pragma unroll
        for (int i = 0; i < 2; ++i) {
            int linear = tid + i * 256;
            int row    = linear >> 3;
            int c4     = (linear & 7) * 4;
            int grow   = tile_m + row;
            if (grow >= M) grow = M - 1;
            const float4 v = *(const float4*)(A + (size_t)grow * K + k0 + c4);
            float* p = &lds_a[row * LDA + c4];
            p[0] = v.x; p[1] = v.y; p[2] = v.z; p[3] = v.w;
        }
        if (full_n) {
            #pragma unroll
            for (int i = 0; i < 4; ++i) {
                int linear = tid + i * 256;
                int krow   = linear >> 5;
                int c4     = (linear & 31) * 4;
                const float4 v = *(const float4*)(B + (size_t)(k0 + krow) * N + tile_n + c4);
                float* p = &lds_b[krow * LDB + c4];
                p[0] = v.x; p[1] = v.y; p[2] = v.z; p[3] = v.w;
            }
        } else {
            #pragma unroll
            for (int i = 0; i < 4; ++i) {
                int linear = tid + i * 256;
                int krow   = linear >> 5;
                int c4     = (linear & 31) * 4;
                float* p = &lds_b[krow * LDB + c4];
                #pragma unroll
                for (int j = 0; j < 4; ++j) {
                    int col = tile_n + c4 + j;
                    p[j] = (col < N) ? B[(size_t)(k0 + krow) * N + col] : 0.0f;
                }
            }
        }
        __syncthreads();
        #pragma unroll
        for (int kk = 0; kk < KCHUNK; kk += 4) {
            const float* ap = &lds_a[(m_sub * 16 + l15) * LDA + kk + half * 2];
            v2f a; a.x = ap[0]; a.y = ap[1];
            const int kb = kk + half * 2;
            #pragma unroll
            for (int t = 0; t < 4; ++t) {
                const float* bp = &lds_b[kb * LDB + wn_l + t * 16 + l15];
                v2f b; b.x = bp[0]; b.y = bp[LDB];
                acc[t] = __builtin_amdgcn_wmma_f32_16x16x4_f32(
                    false, a, false, b, (short)0, acc[t], false, false);
            }
        }
        __syncthreads();
    }
#endif

    // C/D layout: VGPR r -> M = r (lanes 0-15) / r+8 (lanes 16-31), N = lane&15
    #pragma unroll
    for (int t = 0; t < 4; ++t) {
        #pragma unroll
        for (int r = 0; r < 8; ++r) {
            int row = wm + r + half * 8;
            int col = tile_n + wn_l + t * 16 + l15;
            if (row < M && col < N) C[(size_t)row * N + col] = acc[t][r];
        }
    }
}

// ---------------------------------------------------------------------------
// el[n,h] = dot(fs[n,h,:], al[h,:]); er[n,h] (n < Nd) with ar. Wave per (n,h).
// ---------------------------------------------------------------------------
__global__ __launch_bounds__(256)
void spgat_attn_coef(const float* __restrict__ fs, const float* __restrict__ al,
                     const float* __restrict__ ar, float* __restrict__ el,
                     float* __restrict__ er, int Ns, int Nd, int H, int D) {
    int gw   = blockIdx.x * 8 + (threadIdx.x >> 5);
    int lane = threadIdx.x & 31;
    int n = gw / H, h = gw - n * H;
    if (n >= Ns) return;
    const float* frow = fs + (size_t)n * H * D + (size_t)h * D;
    const float* alh  = al + (size_t)h * D;
    const float* arh  = ar + (size_t)h * D;
    float sl = 0.0f, sr = 0.0f;
    for (int d = lane; d < D; d += 32) {
        float f = frow[d];
        sl += f * alh[d];
        sr += f * arh[d];
    }
    #pragma unroll
    for (int off = 16; off > 0; off >>= 1) {
        sl += __shfl_xor(sl, off, 32);
        sr += __shfl_xor(sr, off, 32);
    }
    if (lane == 0) {
        el[n * H + h] = sl;
        if (n < Nd) er[n * H + h] = sr;
    }
}

// order-preserving float <-> u32 encoding for atomic max
__device__ __forceinline__ unsigned ord_enc(float f) {
    unsigned u = __float_as_uint(f);
    return (u & 0x80000000u) ? ~u : (u | 0x80000000u);
}
__device__ __forceinline__ float ord_dec(unsigned u) {
    unsigned i = (u & 0x80000000u) ? (u & 0x7FFFFFFFu) : ~u;
    return __uint_as_float(i);
}

// leaky-relu edge logits + segment max (encoded u32 atomicMax), thread per (e,h)
__global__ void spgat_edge_logit_max(const int* __restrict__ src, const int* __restrict__ dst,
                                     const float* __restrict__ el, const float* __restrict__ er,
                                     float* __restrict__ ebuf, unsigned* __restrict__ emaxu,
                                     int E, int H) {
    int t = blockIdx.x * blockDim.x + threadIdx.x;
    if (t >= E * H) return;
    int e = t / H, h = t - e * H;
    int s = src[e], d = dst[e];
    float v = el[s * H + h] + er[d * H + h];
    v = (v >= 0.0f) ? v : 0.2f * v;           // leaky_relu slope 0.2
    ebuf[t] = v;
    atomicMax(&emaxu[d * H + h], ord_enc(v));
}

// ee = exp(e - max); den += ee (in-place over ebuf), thread per (e,h)
__global__ void spgat_edge_exp_sum(const int* __restrict__ dst, float* __restrict__ ebuf,
                                   const unsigned* __restrict__ emaxu, float* __restrict__ den,
                                   int E, int H) {
    int t = blockIdx.x * blockDim.x + threadIdx.x;
    if (t >= E * H) return;
    int e = t / H, h = t - e * H;
    int d = dst[e];
    float m  = ord_dec(emaxu[d * H + h]);     // segment non-empty (this edge is in it)
    float ee = expf(ebuf[t] - m);
    ebuf[t] = ee;
    atomicAdd(&den[d * H + h], ee);
}

// out[dst,h,:] += alpha * fs[src,h,:], wave per (e,h), coalesced fp32 atomics
__global__ __launch_bounds__(256)
void spgat_edge_aggregate(const int* __restrict__ src, const int* __restrict__ dst,
                          const float* __restrict__ fs, const float* __restrict__ ebuf,
                          const float* __restrict__ den, float* __restrict__ out,
                          int E, int H, int D) {
    int gw   = blockIdx.x * 8 + (threadIdx.x >> 5);
    int lane = threadIdx.x & 31;
    if (gw >= E * H) return;
    int e = gw / H, h = gw - e * H;
    int s = src[e], d = dst[e];
    float alpha = ebuf[gw] / fmaxf(den[d * H + h], 1e-9f);
    const float* frow = fs  + (size_t)s * H * D + (size_t)h * D;
    float*       orow = out + (size_t)d * H * D + (size_t)h * D;
    for (int j = lane; j < D; j += 32)
        atomicAdd(&orow[j], alpha * frow[j]);
}

// x = act ? elu(x + b) : (x + b), in place
__global__ void spgat_bias_act(float* __restrict__ x, const float* __restrict__ b,
                               int total, int N, int act) {
    int t = blockIdx.x * blockDim.x + threadIdx.x;
    if (t >= total) return;
    float v = x[t] + b[t % N];
    if (act) v = (v > 0.0f) ? v : expm1f(v);
    x[t] = v;
}

// ---------------------------------------------------------------------------
// Host side
// ---------------------------------------------------------------------------
static void spgat_run_layer(const float* h_src, const float* W, const float* b,
                            const float* al, const float* ar,
                            const int* src, const int* dst,
                            int Ns, int Nd, int E, int H, int D, int Kdim, int act,
                            float* fs, float* el, float* er, float* ebuf,
                            unsigned* emaxu, float* den, float* out,
                            hipStream_t stream) {
    const int N = H * D;
    // 1) fs = h_src @ W
    dim3 g((Ns + TILE_M - 1) / TILE_M, (N + TILE_N - 1) / TILE_N);
    spgat_gemm_wmma<<<g, 256, 0, stream>>>(h_src, W, fs, Ns, Kdim, N);
    // 2) el / er
    int nwaves = Ns * H;
    spgat_attn_coef<<<(nwaves + 7) / 8, 256, 0, stream>>>(fs, al, ar, el, er, Ns, Nd, H, D);
    // 3) clear accumulation targets
    (void)hipMemsetAsync(emaxu, 0, (size_t)Nd * H * sizeof(unsigned), stream);
    (void)hipMemsetAsync(den,   0, (size_t)Nd * H * sizeof(float),    stream);
    (void)hipMemsetAsync(out,   0, (size_t)Nd * N * sizeof(float),    stream);
    // 4) softmax passes + aggregation
    int EH = E * H;
    spgat_edge_logit_max<<<(EH + 255) / 256, 256, 0, stream>>>(src, dst, el, er, ebuf, emaxu, E, H);
    spgat_edge_exp_sum  <<<(EH + 255) / 256, 256, 0, stream>>>(dst, ebuf, emaxu, den, E, H);
    spgat_edge_aggregate<<<(EH + 7) / 8,     256, 0, stream>>>(src, dst, fs, ebuf, den, out, E, H, D);
    // 5) bias (+ activation)
    int total = Nd * N;
    spgat_bias_act<<<(total + 255) / 256, 256, 0, stream>>>(out, b, total, N, act);
}

extern "C" void kernel_launch(void* const* d_in, const int* in_sizes, int n_in,
                              void* d_out, int out_size, void* d_ws, size_t ws_size,
                              hipStream_t stream) {
    (void)in_sizes; (void)n_in; (void)out_size; (void)ws_size;

    // graph sizes (match reference)
    const int N0 = 100000, N1 = 20000, N2 = 4096, N3 = 1024;
    const int E0 = 300000, E1 = 40960, E2 = 5120;
    const int IN = 256, HID = 128, CLS = 47;
    const int H0 = 4, H1 = 4, H2 = 1;

    const float* x   = (const float*)d_in[0];
    const int* src0  = (const int*)d_in[1];
    const int* dst0  = (const int*)d_in[2];
    const int* src1  = (const int*)d_in[3];
    const int* dst1  = (const int*)d_in[4];
    const int* src2  = (const int*)d_in[5];
    const int* dst2  = (const int*)d_in[6];
    const float* W0  = (const float*)d_in[7];
    const float* b0  = (const float*)d_in[8];
    const float* al0 = (const float*)d_in[9];
    const float* ar0 = (const float*)d_in[10];
    const float* W1  = (const float*)d_in[11];
    const float* b1  = (const float*)d_in[12];
    const float* al1 = (const float*)d_in[13];
    const float* ar1 = (const float*)d_in[14];
    const float* W2  = (const float*)d_in[15];
    const float* b2  = (const float*)d_in[16];
    const float* al2 = (const float*)d_in[17];
    const float* ar2 = (const float*)d_in[18];

    float* out = (float*)d_out;   // [N3, CLS]
    float* ws  = (float*)d_ws;

    // workspace layout (floats)
    size_t o = 0;
    float* fs0 = ws + o; o += (size_t)N0 * H0 * HID;   // 51,200,000
    float* h1  = ws + o; o += (size_t)N1 * H0 * HID;   // 10,240,000
    float* fs1 = ws + o; o += (size_t)N1 * H1 * HID;   // 10,240,000
    float* h2  = ws + o; o += (size_t)N2 * H1 * HID;   //  2,097,152
    float* fs2 = ws + o; o += (size_t)N2 * CLS;        //    192,512
    float* ebuf = ws + o; o += (size_t)E0 * H0;        //  1,200,000
    float* el  = ws + o; o += (size_t)N0 * H0;         //    400,000
    float* er  = ws + o; o += (size_t)N1 * H0;         //     80,000
    unsigned* emaxu = (unsigned*)(ws + o); o += (size_t)N1 * H0; // 80,000
    float* den = ws + o; o += (size_t)N1 * H0;         //     80,000

    // Layer 0: [N0,IN] -> [N1, H0*HID], ELU
    spgat_run_layer(x,  W0, b0, al0, ar0, src0, dst0,
                    N0, N1, E0, H0, HID, IN, 1,
                    fs0, el, er, ebuf, emaxu, den, h1, stream);
    // Layer 1: [N1, H0*HID] -> [N2, H1*HID], ELU
    spgat_run_layer(h1, W1, b1, al1, ar1, src1, dst1,
                    N1, N2, E1, H1, HID, H0 * HID, 1,
                    fs1, el, er, ebuf, emaxu, den, h2, stream);
    // Layer 2: [N2, H1*HID] -> [N3, CLS], no act; H2==1 so mean == identity.
    // Aggregate directly into d_out.
    spgat_run_layer(h2, W2, b2, al2, ar2, src2, dst2,
                    N2, N3, E2, H2, CLS, H1 * HID, 0,
                    fs2, el, er, ebuf, emaxu, den, out, stream);
    (void)fs2;
}